// DenseFlashAttention_85607288144368
// MI455X (gfx1250) — compile-verified
//
#include <hip/hip_runtime.h>
#include <cstdint>
#include <cstddef>

#define FDIM 64
#define HDIM 4

typedef __attribute__((ext_vector_type(2))) float v2f;
typedef __attribute__((ext_vector_type(8))) float v8f;
typedef __attribute__((ext_vector_type(4))) unsigned int v4u;
typedef __attribute__((ext_vector_type(4))) int v4i;
typedef __attribute__((ext_vector_type(8))) int v8i;

__device__ __forceinline__ float softplusf(float v) {
    return v > 20.f ? v : log1pf(expf(v));
}

// float atomic max via int punning (requires init to -inf)
__device__ __forceinline__ void atomicMaxFloat(float* addr, float val) {
    if (val >= 0.f) atomicMax((int*)addr, __float_as_int(val));
    else            atomicMin((unsigned int*)addr, __float_as_uint(val));
}

// ---------------------------------------------------------------------------
// v_r[h,f] = sum_g w_proj[h,f,g]*radial_score[h,g]; v_t likewise.
__global__ __launch_bounds__(256) void combine_kernel(
    const float* __restrict__ w_proj, const float* __restrict__ rs,
    const float* __restrict__ ts, float* __restrict__ v_r, float* __restrict__ v_t) {
    const int t = threadIdx.x;          // 256 == H*F
    const int h = t >> 6, f = t & 63;
    float ar = 0.f, at = 0.f;
    const float* wrow = w_proj + ((size_t)h * FDIM + f) * FDIM;
    for (int g = 0; g < FDIM; ++g) {
        float w = wrow[g];
        ar += w * rs[h * FDIM + g];
        at += w * ts[h * FDIM + g];
    }
    v_r[t] = ar; v_t[t] = at;
}

// ---------------------------------------------------------------------------
__global__ void init_kernel(float* __restrict__ acc,
                            float* __restrict__ z_r, float* __restrict__ z_t,
                            float* __restrict__ sa_r, float* __restrict__ sa_t,
                            float* __restrict__ m_r, float* __restrict__ m_t,
                            size_t nAcc, size_t nNH) {
    size_t tid = (size_t)blockIdx.x * blockDim.x + threadIdx.x;
    size_t stride = (size_t)gridDim.x * blockDim.x;
    for (size_t i = tid; i < nAcc; i += stride) acc[i] = 0.f;
    for (size_t i = tid; i < nNH; i += stride) {
        z_r[i] = 0.f; z_t[i] = 0.f; sa_r[i] = 0.f; sa_t[i] = 0.f;
        m_r[i] = -__builtin_inff(); m_t[i] = -__builtin_inff();
    }
}

// ---------------------------------------------------------------------------
// TDM stage: load a [rows<=16] x 64 fp32 tile starting at `gsrc` into LDS at
// byte offset lds_off, with 2-DWORD padding after every 64 DWORDs
// (row stride 66 floats). Rows >= rows_valid are zero-filled by the TDM OOB
// rule (tensor_dim1 = rows_valid).
__device__ __forceinline__ void tdm_load_tile(const float* gsrc,
                                              unsigned int lds_off,
                                              unsigned int rows_valid) {
    unsigned long long ga = (unsigned long long)(const void*)gsrc;
    v4u g0;
    g0.x = 1u;                                   // count=1 (valid descriptor)
    g0.y = lds_off;                              // lds_addr (bytes)
    g0.z = (unsigned int)ga;                     // global_addr[31:0]
    g0.w = (unsigned int)((ga >> 32) & 0x1FFFFFFu) | (2u << 30); // [56:32], type=2

    const unsigned int tensor_dim0 = 64u;        // line length (elements)
    const unsigned int tensor_dim1 = rows_valid; // OOB rows -> zero
    const unsigned int tile_dim0   = 64u;
    const unsigned int tile_dim1   = 16u;
    const unsigned long long stride0 = 64ull;    // elements between lines

    v8i g1;
    // data_size=2 (4B) | pad_enable | pad_interval=5 (64 DW) | pad_amount=1 (2 DW)
    g1[0] = (int)((2u << 16) | (1u << 20) | (5u << 22) | (1u << 25));
    g1[1] = (int)((tensor_dim0 & 0xFFFFu) << 16);          // abar_addr=0
    g1[2] = (int)((tensor_dim0 >> 16) | ((tensor_dim1 & 0xFFFFu) << 16));
    g1[3] = (int)((tensor_dim1 >> 16) | (tile_dim0 << 16));
    g1[4] = (int)(tile_dim1);                              // tile_dim2 = 0
    g1[5] = (int)(unsigned int)stride0;                    // stride0[31:0]
    g1[6] = (int)(unsigned int)(stride0 >> 32);            // stride0[47:32], stride1 lo=0
    g1[7] = 0;                                             // stride1 hi

    v4i g2 = {0, 0, 0, 0};
    v4i g3 = {0, 0, 0, 0};
#if defined(__clang_major__) && (__clang_major__ >= 23)
    v8i g4 = {0, 0, 0, 0, 0, 0, 0, 0};
    __builtin_amdgcn_tensor_load_to_lds(g0, g1, g2, g3, g4, 0);
#else
    __builtin_amdgcn_tensor_load_to_lds(g0, g1, g2, g3, 0);
#endif
}

// ---------------------------------------------------------------------------
// Projections radial/tang [H,N,F] via fp32 WMMA, plus sr/st = x . v_{r,t}.
__global__ __launch_bounds__(256) void proj_kernel(
    const float* __restrict__ x,
    const float* __restrict__ radial_w, const float* __restrict__ tangential_w,
    const float* __restrict__ v_r, const float* __restrict__ v_t,
    float* __restrict__ radial_proj, float* __restrict__ tang_proj,
    float* __restrict__ sr, float* __restrict__ st, int N) {
    __shared__ float ldsX[16 * 66];     // stride 66 -> conflict free
    const int t = threadIdx.x;
    const int node0 = blockIdx.x * 16;

    // stage 16x64 tile of x via Tensor Data Mover (zero-pads rows >= N)
    if (t < 32) {
        unsigned int lds_off = (unsigned int)(uintptr_t)(&ldsX[0]);
        unsigned int rows_valid = (unsigned int)(N - node0);
        if (rows_valid > 16u) rows_valid = 16u;
        tdm_load_tile(x + (size_t)node0 * FDIM, lds_off, rows_valid);
        __builtin_amdgcn_s_wait_tensorcnt(0);
    }
    __syncthreads();

    // sr/st: 16 rows x 4 heads x 2 types = 128 dot products (waves 0-3)
    if (t < 128) {
        int row = t & 15, h = (t >> 4) & 3, ty = t >> 6;
        int node = node0 + row;
        if (node < N) {
            const float* v = (ty == 0 ? v_r : v_t) + h * FDIM;
            float a = 0.f;
            for (int f = 0; f < FDIM; ++f) a += ldsX[row * 66 + f] * v[f];
            (ty == 0 ? sr : st)[(size_t)h * N + node] = a;
        }
    }

    const int wave = t >> 5, lane = t & 31;
    const int lrow = lane & 15, lhi = lane >> 4;
    // 32 wave-tasks: {matrix 0/1} x {head 0..3} x {coltile 0..3}, 8 waves
    for (int task = wave; task < 32; task += 8) {
        const int m = task >> 4;
        const int h = (task >> 2) & 3;
        const int c = task & 3;
        const float* W = (m == 0 ? radial_w : tangential_w) + (size_t)h * FDIM * FDIM;
        const int ncol = c * 16 + lrow;
        v8f acc = {};
        for (int ks = 0; ks < 16; ++ks) {
            const int kb = 4 * ks + 2 * lhi;
            v2f a, b;
            a.x = ldsX[lrow * 66 + kb];
            a.y = ldsX[lrow * 66 + kb + 1];
            b.x = W[(size_t)kb * FDIM + ncol];
            b.y = W[(size_t)(kb + 1) * FDIM + ncol];
            acc = __builtin_amdgcn_wmma_f32_16x16x4_f32(
                false, a, false, b, (short)0, acc, false, false);
        }
        float* dst = (m == 0 ? radial_proj : tang_proj) + (size_t)h * N * FDIM;
        for (int v = 0; v < 8; ++v) {
            const int node = node0 + v + 8 * lhi;
            if (node < N) dst[(size_t)node * FDIM + ncol] = acc[v];
        }
    }
}

// ---------------------------------------------------------------------------
// Edge logits + running per-receiver max. er/et layout [E,H].
__global__ __launch_bounds__(256) void logits_kernel(
    const int* __restrict__ ei, const float* __restrict__ edge_len,
    const float* __restrict__ sr, const float* __restrict__ st,
    const float* __restrict__ lsc, const float* __restrict__ tb,
    const float* __restrict__ tw,
    float* __restrict__ er, float* __restrict__ et,
    float* __restrict__ m_r, float* __restrict__ m_t, int N, int E) {
    size_t tid = (size_t)blockIdx.x * blockDim.x + threadIdx.x;
    if (tid >= (size_t)E * HDIM) return;
    const int h = (int)(tid & 3);
    const size_t e = tid >> 2;
    const int s = ei[e], r = ei[(size_t)E + e];
    const float len = edge_len[e];
    const float scale = softplusf(lsc[0]);
    float rl = (sr[(size_t)h * N + s] - sr[(size_t)h * N + r]) - scale * len;
    const float temp = softplusf(tb[h] + tw[h] * len);
    rl = rl / (temp + 1e-4f);
    const float tl = st[(size_t)h * N + s] - st[(size_t)h * N + r];
    er[tid] = rl; et[tid] = tl;
    atomicMaxFloat(&m_r[(size_t)r * HDIM + h], rl);
    atomicMaxFloat(&m_t[(size_t)r * HDIM + h], tl);
}

// ---------------------------------------------------------------------------
__global__ __launch_bounds__(256) void expsum_kernel(
    const int* __restrict__ ei, float* __restrict__ er, float* __restrict__ et,
    const float* __restrict__ m_r, const float* __restrict__ m_t,
    float* __restrict__ z_r, float* __restrict__ z_t, int E) {
    size_t tid = (size_t)blockIdx.x * blockDim.x + threadIdx.x;
    if (tid >= (size_t)E * HDIM) return;
    const int h = (int)(tid & 3);
    const size_t e = tid >> 2;
    const int r = ei[(size_t)E + e];
    const float a = expf(er[tid] - m_r[(size_t)r * HDIM + h]);
    const float b = expf(et[tid] - m_t[(size_t)r * HDIM + h]);
    er[tid] = a; et[tid] = b;
    unsafeAtomicAdd(&z_r[(size_t)r * HDIM + h], a);
    unsafeAtomicAdd(&z_t[(size_t)r * HDIM + h], b);
}

// ---------------------------------------------------------------------------
// Scatter messages: acc[n,h,f] += a_r*radial[h,s,f] + a_t*tang[h,s,f]; sa sums.
__global__ __launch_bounds__(256) void message_kernel(
    const int* __restrict__ ei,
    const float* __restrict__ er, const float* __restrict__ et,
    const float* __restrict__ z_r, const float* __restrict__ z_t,
    const float* __restrict__ radial_proj, const float* __restrict__ tang_proj,
    float* __restrict__ acc, float* __restrict__ sa_r, float* __restrict__ sa_t,
    int N, int E) {
    const int h = threadIdx.x >> 6;
    const int f = threadIdx.x & 63;
    const size_t e0 = (size_t)blockIdx.x * 8;
    for (int i = 0; i < 8; ++i) {
        const size_t e = e0 + i;
        if (e >= (size_t)E) break;
        const int s = ei[e], r = ei[(size_t)E + e];
        const float ar = er[e * HDIM + h] / z_r[(size_t)r * HDIM + h];
        const float at = et[e * HDIM + h] / z_t[(size_t)r * HDIM + h];
        const float val = ar * radial_proj[((size_t)h * N + s) * FDIM + f]
                        + at * tang_proj[((size_t)h * N + s) * FDIM + f];
        unsafeAtomicAdd(&acc[((size_t)r * HDIM + h) * FDIM + f], val);
        if (f == 0) {
            unsafeAtomicAdd(&sa_r[(size_t)r * HDIM + h], ar);
            unsafeAtomicAdd(&sa_t[(size_t)r * HDIM + h], at);
        }
    }
}

// ---------------------------------------------------------------------------
// tmp = nan_to_num(mean_h(acc - radial*sa_r - tang*sa_t)); out = x + tmp@w_out
__global__ __launch_bounds__(256) void final_kernel(
    const float* __restrict__ x, const float* __restrict__ w_out,
    const float* __restrict__ radial_proj, const float* __restrict__ tang_proj,
    const float* __restrict__ acc_buf,
    const float* __restrict__ sa_r, const float* __restrict__ sa_t,
    float* __restrict__ out, int N) {
    __shared__ float ldsT[16 * 66];
    const int t = threadIdx.x;
    const int node0 = blockIdx.x * 16;
    for (int i = 0; i < 4; ++i) {
        int idx = t + 256 * i;
        int row = idx >> 6, col = idx & 63;
        int node = node0 + row;
        float v = 0.f;
        if (node < N) {
            float s = 0.f;
            for (int h = 0; h < HDIM; ++h) {
                float a  = acc_buf[((size_t)node * HDIM + h) * FDIM + col];
                float pr = radial_proj[((size_t)h * N + node) * FDIM + col];
                float pt = tang_proj[((size_t)h * N + node) * FDIM + col];
                s += a - pr * sa_r[(size_t)node * HDIM + h]
                       - pt * sa_t[(size_t)node * HDIM + h];
            }
            v = s * 0.25f;
            if (!(v == v)) v = 0.f;                       // NaN -> 0
            else if (v >  3.4028234e38f) v =  3.4028234e38f;
            else if (v < -3.4028234e38f) v = -3.4028234e38f;
        }
        ldsT[row * 66 + col] = v;
    }
    __syncthreads();
    const int wave = t >> 5, lane = t & 31;
    const int lrow = lane & 15, lhi = lane >> 4;
    if (wave < 4) {
        const int ncol = wave * 16 + lrow;
        v8f d = {};
        for (int ks = 0; ks < 16; ++ks) {
            const int kb = 4 * ks + 2 * lhi;
            v2f a, b;
            a.x = ldsT[lrow * 66 + kb];
            a.y = ldsT[lrow * 66 + kb + 1];
            b.x = w_out[(size_t)kb * FDIM + ncol];
            b.y = w_out[(size_t)(kb + 1) * FDIM + ncol];
            d = __builtin_amdgcn_wmma_f32_16x16x4_f32(
                false, a, false, b, (short)0, d, false, false);
        }
        for (int v = 0; v < 8; ++v) {
            const int node = node0 + v + 8 * lhi;
            if (node < N)
                out[(size_t)node * FDIM + ncol] =
                    x[(size_t)node * FDIM + ncol] + d[v];
        }
    }
}

// ---------------------------------------------------------------------------
extern "C" void kernel_launch(void* const* d_in, const int* in_sizes, int n_in,
                              void* d_out, int out_size, void* d_ws, size_t ws_size,
                              hipStream_t stream) {
    const float* x            = (const float*)d_in[0];
    const int*   ei           = (const int*)d_in[1];
    /* d_in[2] edge_vec unused by reference */
    const float* edge_len     = (const float*)d_in[3];
    const float* w_proj       = (const float*)d_in[4];
    const float* radial_w     = (const float*)d_in[5];
    const float* tangential_w = (const float*)d_in[6];
    const float* w_out        = (const float*)d_in[7];
    const float* rscore       = (const float*)d_in[8];
    const float* tscore       = (const float*)d_in[9];
    const float* lsc          = (const float*)d_in[10];
    const float* tb           = (const float*)d_in[11];
    const float* tw           = (const float*)d_in[12];

    const int N = in_sizes[0] / FDIM;
    const int E = in_sizes[3];

    float* p = (float*)d_ws;
    float* radial_proj = p; p += (size_t)HDIM * N * FDIM;
    float* tang_proj   = p; p += (size_t)HDIM * N * FDIM;
    float* acc         = p; p += (size_t)N * HDIM * FDIM;
    float* sr          = p; p += (size_t)HDIM * N;
    float* st          = p; p += (size_t)HDIM * N;
    float* m_r         = p; p += (size_t)N * HDIM;
    float* m_t         = p; p += (size_t)N * HDIM;
    float* z_r         = p; p += (size_t)N * HDIM;
    float* z_t         = p; p += (size_t)N * HDIM;
    float* sa_r        = p; p += (size_t)N * HDIM;
    float* sa_t        = p; p += (size_t)N * HDIM;
    float* er          = p; p += (size_t)E * HDIM;
    float* et          = p; p += (size_t)E * HDIM;
    float* v_r         = p; p += (size_t)HDIM * FDIM;
    float* v_t         = p; p += (size_t)HDIM * FDIM;

    combine_kernel<<<1, 256, 0, stream>>>(w_proj, rscore, tscore, v_r, v_t);

    const size_t nAcc = (size_t)N * HDIM * FDIM;
    const size_t nNH  = (size_t)N * HDIM;
    init_kernel<<<2048, 256, 0, stream>>>(acc, z_r, z_t, sa_r, sa_t, m_r, m_t,
                                          nAcc, nNH);

    const int nTiles = (N + 15) / 16;
    proj_kernel<<<nTiles, 256, 0, stream>>>(x, radial_w, tangential_w, v_r, v_t,
                                            radial_proj, tang_proj, sr, st, N);

    const size_t nEH = (size_t)E * HDIM;
    const int eBlocks = (int)((nEH + 255) / 256);
    logits_kernel<<<eBlocks, 256, 0, stream>>>(ei, edge_len, sr, st, lsc, tb, tw,
                                               er, et, m_r, m_t, N, E);
    expsum_kernel<<<eBlocks, 256, 0, stream>>>(ei, er, et, m_r, m_t, z_r, z_t, E);

    message_kernel<<<(E + 7) / 8, 256, 0, stream>>>(ei, er, et, z_r, z_t,
                                                    radial_proj, tang_proj,
                                                    acc, sa_r, sa_t, N, E);

    final_kernel<<<nTiles, 256, 0, stream>>>(x, w_out, radial_proj, tang_proj,
                                             acc, sa_r, sa_t, (float*)d_out, N);
}